// GNN_classifier_65652870087075
// MI455X (gfx1250) — compile-verified
//
#include <hip/hip_runtime.h>
#include <hip/hip_bf16.h>

typedef __attribute__((ext_vector_type(2))) float v2f;
typedef __attribute__((ext_vector_type(8))) float v8f;

#define GRID1(c) ((unsigned)(((c) + 255) / 256))

// ---------------- utility kernels ----------------
__global__ void zero_f32(float* __restrict__ p, int count) {
  int i = blockIdx.x * blockDim.x + threadIdx.x;
  if (i < count) p[i] = 0.0f;
}

__global__ void set_u32(unsigned* __restrict__ p, unsigned v, int count) {
  int i = blockIdx.x * blockDim.x + threadIdx.x;
  if (i < count) p[i] = v;
}

// ---------------- degree / norm ----------------
__global__ void degree_kernel(const int* __restrict__ dst, float* __restrict__ deg, int nE) {
  int i = blockIdx.x * blockDim.x + threadIdx.x;
  if (i < nE) atomicAdd(&deg[dst[i]], 1.0f);
}

__global__ void norm_kernel(const float* __restrict__ deg, float* __restrict__ nrm, int nN) {
  int i = blockIdx.x * blockDim.x + threadIdx.x;
  if (i < nN) nrm[i] = rsqrtf(fmaxf(deg[i], 1.0f));
}

// ---------------- edge propagation: y[dst] += x[src]*norm[src] ----------------
// one wave32 per edge; 32 lanes x float2 = one 64-float node row
__global__ __launch_bounds__(256) void scatter_edges(
    const float* __restrict__ x, const float* __restrict__ nrm,
    const int* __restrict__ src, const int* __restrict__ dst,
    float* __restrict__ y, int nE) {
  int gw = (int)((blockIdx.x * blockDim.x + threadIdx.x) >> 5);
  int lane = threadIdx.x & 31;
  if (gw >= nE) return;
  int s = src[gw];
  int d = dst[gw];
  float ns = nrm[s];
  float2 v = *reinterpret_cast<const float2*>(x + (size_t)s * 64 + lane * 2);
  float* yr = y + (size_t)d * 64 + lane * 2;
  atomicAdd(yr, v.x * ns);
  atomicAdd(yr + 1, v.y * ns);
}

// X1 = -norm * y ; y cleared for next propagation
__global__ void combine_k1(float* __restrict__ y, const float* __restrict__ nrm,
                           float* __restrict__ xout, int total) {
  int i = blockIdx.x * blockDim.x + threadIdx.x;
  if (i >= total) return;
  int n = i >> 6;
  xout[i] = -nrm[n] * y[i];
  y[i] = 0.0f;
}

// Xk = -2*norm*y - Xprev2 ; y cleared for next propagation
__global__ void combine_k(float* __restrict__ y, const float* __restrict__ nrm,
                          const float* __restrict__ xp2, float* __restrict__ xout, int total) {
  int i = blockIdx.x * blockDim.x + threadIdx.x;
  if (i >= total) return;
  int n = i >> 6;
  xout[i] = -2.0f * nrm[n] * y[i] - xp2[i];
  y[i] = 0.0f;
}

__global__ void bias_relu(float* __restrict__ h, const float* __restrict__ b, int total) {
  int i = blockIdx.x * blockDim.x + threadIdx.x;
  if (i >= total) return;
  float v = h[i] + b[i & 63];
  h[i] = v > 0.0f ? v : 0.0f;
}

// ---------------- WMMA f32 GEMM: acc[N x 64] += T[N x 64] @ Wk[64 x 64] ----------------
// block = 256 threads = 8 waves; block tile 32 rows x 64 cols; one 16x16 tile per wave.
// V_WMMA_F32_16X16X4_F32 layouts (CDNA5 ISA):
//   A 16x4: lane-half selects K-half {0,1}|{2,3}; VGPR selects K within half; M = lane&15
//   B 4x16: VGPR r -> row K = r (lanes 0-15) / K = r+2 (lanes 16-31); N = lane&15
//   C/D 16x16: VGPR r -> M = r (+8 for upper lanes), N = lane&15
__global__ __launch_bounds__(256) void cheb_gemm16(
    const float* __restrict__ T, const float* __restrict__ Wk,
    float* __restrict__ acc, int nrows) {
  __shared__ float sA[32 * 64];  // 8 KB
  __shared__ float sB[64 * 64];  // 16 KB
  const int tid = threadIdx.x;
  const int m0 = blockIdx.x * 32;

  for (int i = tid; i < 32 * 64; i += 256) {
    int row = m0 + (i >> 6);
    sA[i] = (row < nrows) ? T[(size_t)row * 64 + (i & 63)] : 0.0f;
  }
  for (int i = tid; i < 64 * 64; i += 256) sB[i] = Wk[i];
  __syncthreads();

  const int wave  = tid >> 5;
  const int lane  = tid & 31;
  const int mBase = (wave >> 2) * 16;  // 0 or 16
  const int n0    = (wave & 3) * 16;   // 0,16,32,48
  const int lhalf = lane >> 4;
  const int l16   = lane & 15;

  v8f c;
  const int growBase = m0 + mBase + lhalf * 8;
  const int gcol = n0 + l16;
#pragma unroll
  for (int r = 0; r < 8; ++r) {
    int row = growBase + r;
    c[r] = (row < nrows) ? acc[(size_t)row * 64 + gcol] : 0.0f;
  }

  const float* aRow = &sA[(mBase + l16) * 64];
#pragma unroll
  for (int kk = 0; kk < 64; kk += 4) {
    const int ka = kk + lhalf * 2;
    v2f a, b;
    a.x = aRow[ka];
    a.y = aRow[ka + 1];
    b.x = sB[(ka + 0) * 64 + n0 + l16];
    b.y = sB[(ka + 1) * 64 + n0 + l16];
    c = __builtin_amdgcn_wmma_f32_16x16x4_f32(false, a, false, b, (short)0, c, false, false);
  }

#pragma unroll
  for (int r = 0; r < 8; ++r) {
    int row = growBase + r;
    if (row < nrows) acc[(size_t)row * 64 + gcol] = c[r];
  }
}

// ---------------- attention pooling ----------------
__device__ __forceinline__ unsigned f2key(float f) {
  int b = __float_as_int(f);
  return (b < 0) ? ~(unsigned)b : ((unsigned)b | 0x80000000u);
}
__device__ __forceinline__ float key2f(unsigned u) {
  int b = (u & 0x80000000u) ? (int)(u ^ 0x80000000u) : ~(int)u;
  return __int_as_float(b);
}

// one wave per node: gate[n] = h[n].Wg + bg ; per-graph max via order-preserving u32 key
__global__ __launch_bounds__(256) void gate_kernel(
    const float* __restrict__ h, const float* __restrict__ Wg, const float* __restrict__ bg,
    const int* __restrict__ seg, float* __restrict__ gate, unsigned* __restrict__ umax, int nN) {
  int gw = (int)((blockIdx.x * blockDim.x + threadIdx.x) >> 5);
  int lane = threadIdx.x & 31;
  if (gw >= nN) return;
  float2 hv = *reinterpret_cast<const float2*>(h + (size_t)gw * 64 + lane * 2);
  float2 wv = *reinterpret_cast<const float2*>(Wg + lane * 2);
  float s = hv.x * wv.x + hv.y * wv.y;
#pragma unroll
  for (int o = 16; o > 0; o >>= 1) s += __shfl_down(s, o, 32);
  if (lane == 0) {
    s += bg[0];
    gate[gw] = s;
    atomicMax(&umax[seg[gw]], f2key(s));
  }
}

__global__ void expsum_kernel(float* __restrict__ gate, const unsigned* __restrict__ umax,
                              const int* __restrict__ seg, float* __restrict__ z, int nN) {
  int n = blockIdx.x * blockDim.x + threadIdx.x;
  if (n >= nN) return;
  int g = seg[n];
  float e = expf(gate[n] - key2f(umax[g]));
  gate[n] = e;
  atomicAdd(&z[g], e);
}

// hg[g] += h[n]*(e[n]/z[g]) ; LDS-staged per-block partials (16 graphs x 64 dims)
__global__ __launch_bounds__(256) void pool_kernel(
    const float* __restrict__ h, const float* __restrict__ e, const float* __restrict__ z,
    const int* __restrict__ seg, float* __restrict__ hg, int nN) {
  __shared__ float sh[16 * 64];
  int tid = threadIdx.x;
  for (int i = tid; i < 16 * 64; i += 256) sh[i] = 0.0f;
  __syncthreads();
  int n = blockIdx.x * 256 + tid;
  if (n < nN) {
    int g = seg[n];
    float alpha = e[n] / z[g];
    const float* hr = h + (size_t)n * 64;
    float* dstp = &sh[g * 64];
#pragma unroll 8
    for (int d = 0; d < 64; ++d) atomicAdd(&dstp[d], hr[d] * alpha);
  }
  __syncthreads();
  for (int i = tid; i < 16 * 64; i += 256)
    if (sh[i] != 0.0f) atomicAdd(&hg[i], sh[i]);
}

// final MLP: out[16x3] = (hg[16x64]@W1 + b1) @ W2 + b2, single block
__global__ __launch_bounds__(1024) void mlp_kernel(
    const float* __restrict__ hg, const float* __restrict__ W1, const float* __restrict__ b1,
    const float* __restrict__ W2, const float* __restrict__ b2, float* __restrict__ out) {
  __shared__ float a2[16 * 64];
  int tid = threadIdx.x;
  int g = tid >> 6, j = tid & 63;
  float s = b1[j];
  const float* hr = hg + g * 64;
#pragma unroll 8
  for (int d = 0; d < 64; ++d) s += hr[d] * W1[d * 64 + j];
  a2[tid] = s;
  __syncthreads();
  if (tid < 48) {
    int gg = tid / 3, cc = tid % 3;
    float o = b2[cc];
    const float* ar = &a2[gg * 64];
#pragma unroll 8
    for (int jj = 0; jj < 64; ++jj) o += ar[jj] * W2[jj * 3 + cc];
    out[gg * 3 + cc] = o;
  }
}

// ---------------- host ----------------
extern "C" void kernel_launch(void* const* d_in, const int* in_sizes, int n_in,
                              void* d_out, int out_size, void* d_ws, size_t ws_size,
                              hipStream_t stream) {
  const float* feat = (const float*)d_in[0];
  const float* Wc   = (const float*)d_in[1];
  const float* bc   = (const float*)d_in[2];
  const float* Wg   = (const float*)d_in[3];
  const float* bg   = (const float*)d_in[4];
  const float* W1   = (const float*)d_in[5];
  const float* b1   = (const float*)d_in[6];
  const float* W2   = (const float*)d_in[7];
  const float* b2   = (const float*)d_in[8];
  const int*   src  = (const int*)d_in[9];
  const int*   dst  = (const int*)d_in[10];
  const int*   seg  = (const int*)d_in[11];

  const int N  = in_sizes[0] / 64;
  const int E  = in_sizes[9];
  const int L  = in_sizes[2] / 64;             // layers
  const int K  = in_sizes[1] / (L * 64 * 64);  // Chebyshev order
  const int NN = N * 64;

  float* P = (float*)d_ws;
  float* nrm  = P;  P += N;
  float* deg  = P;  P += N;
  float* Y    = P;  P += NN;
  float* T1   = P;  P += NN;
  float* T2   = P;  P += NN;
  float* T3   = P;  P += NN;
  float* H0   = P;  P += NN;
  float* H1   = P;  P += NN;
  float* gate = P;  P += N;
  float* z    = P;  P += 16;
  float* hg   = P;  P += 16 * 64;
  unsigned* umax = (unsigned*)P;

  // scratch init (d_ws is poisoned before timing; zero everything read-before-write)
  zero_f32<<<GRID1(N), 256, 0, stream>>>(deg, N);
  zero_f32<<<GRID1(NN), 256, 0, stream>>>(Y, NN);
  zero_f32<<<1, 64, 0, stream>>>(z, 16);
  zero_f32<<<1, 1024, 0, stream>>>(hg, 16 * 64);
  set_u32<<<1, 64, 0, stream>>>(umax, 0x00800000u /* key(-FLT_MAX) */, 16);

  degree_kernel<<<GRID1(E), 256, 0, stream>>>(dst, deg, E);
  norm_kernel<<<GRID1(N), 256, 0, stream>>>(deg, nrm, N);

  const unsigned gemmGrid = (unsigned)((N + 31) / 32);
  const unsigned edgeGrid = (unsigned)((E + 7) / 8);

  const float* hin = feat;
  float* hbufs[2] = {H0, H1};
  float* bufs[3]  = {T1, T2, T3};

  for (int l = 0; l < L; ++l) {
    const float* Wl = Wc + (size_t)l * K * 64 * 64;
    float* hout = hbufs[l & 1];
    zero_f32<<<GRID1(NN), 256, 0, stream>>>(hout, NN);
    // k = 0 : T0 = h
    cheb_gemm16<<<gemmGrid, 256, 0, stream>>>(hin, Wl, hout, N);
    // k = 1 : T1 = -P(h)
    scatter_edges<<<edgeGrid, 256, 0, stream>>>(hin, nrm, src, dst, Y, E);
    combine_k1<<<GRID1(NN), 256, 0, stream>>>(Y, nrm, T1, NN);
    cheb_gemm16<<<gemmGrid, 256, 0, stream>>>(T1, Wl + 64 * 64, hout, N);
    const float* xp2 = hin;
    const float* xp1 = T1;
    for (int k = 2; k < K; ++k) {
      float* xk = bufs[(k - 1) % 3];  // rotates over T1,T2,T3; X_{k-3} is dead
      scatter_edges<<<edgeGrid, 256, 0, stream>>>(xp1, nrm, src, dst, Y, E);
      combine_k<<<GRID1(NN), 256, 0, stream>>>(Y, nrm, xp2, xk, NN);
      cheb_gemm16<<<gemmGrid, 256, 0, stream>>>(xk, Wl + (size_t)k * 64 * 64, hout, N);
      xp2 = xp1;
      xp1 = xk;
    }
    bias_relu<<<GRID1(NN), 256, 0, stream>>>(hout, bc + l * 64, NN);
    hin = hout;
  }

  gate_kernel<<<(unsigned)((N + 7) / 8), 256, 0, stream>>>(hin, Wg, bg, seg, gate, umax, N);
  expsum_kernel<<<GRID1(N), 256, 0, stream>>>(gate, umax, seg, z, N);
  pool_kernel<<<GRID1(N), 256, 0, stream>>>(hin, gate, z, seg, hg, N);
  mlp_kernel<<<1, 1024, 0, stream>>>(hg, W1, b1, W2, b2, (float*)d_out);
}